// cknrm_38457137168916
// MI455X (gfx1250) — compile-verified
//
#include <hip/hip_runtime.h>
#include <math.h>

// ---------------------------------------------------------------------------
// Conv-KNRM forward for MI455X (gfx1250, wave32, WMMA + async/TDM).
// All GEMMs (n-gram convolutions and q·dᵀ similarity) run on
// V_WMMA_F32_16X16X4_F32 to preserve fp32 fidelity (bin-0 RBF has sigma=1e-3,
// an exact-match detector that bf16 would destroy).
//
// CDNA5 paths used:
//   * v_wmma_f32_16x16x4_f32 for all GEMMs (independent accumulator chains)
//   * global_load_async_to_lds_b128 + s_wait_asynccnt for the embedding gather
//   * tensor_load_to_lds (TDM, D# with LDS padding) + s_wait_tensorcnt for the
//     64x128 doc-feature tiles in the pooling kernel
//
// Workspace (floats), B=128: qfeat 1.57M, dfeat 25.2M, aq 12K, ad 197K,
// pools 12.7K -> ~108 MB total.
// ---------------------------------------------------------------------------

typedef __attribute__((ext_vector_type(2))) float v2f;
typedef __attribute__((ext_vector_type(8))) float v8f;
typedef __attribute__((ext_vector_type(4))) unsigned int u32x4;
typedef __attribute__((ext_vector_type(8))) unsigned int u32x8;

#define D_EMB 300
#define C_CH  128
#define LQ    32
#define LD    512
#define NB    11
#define W1    (D_EMB * C_CH)   // stride between taps of a weight tensor
#define EPAD  308              // LDS row stride, emb tile (conflict-free, 16B)
#define OPAD  132              // LDS row stride for 128-wide feature rows
#define QPAD  132

// fp32 WMMA: D(16x16) = A(16x4) * B(4x16) + C, f32 accumulate.
// A layout: lane&15 = row M; VGPR j, half h=(lane>>4) -> K = j + 2h.
// C/D layout: VGPR r -> M = r + 8*(lane>>4), N = lane & 15.
static __device__ __forceinline__ v8f wmma4(v2f a, v2f b, v8f c) {
  return __builtin_amdgcn_wmma_f32_16x16x4_f32(false, a, false, b, (short)0, c,
                                               false, false);
}

// Per-lane async global->LDS copy of 16 bytes (ASYNCcnt-tracked).
static __device__ __forceinline__ void async_copy_b128(unsigned lds_off,
                                                       unsigned long long gaddr) {
  asm volatile("global_load_async_to_lds_b128 %0, %1, off"
               :: "v"(lds_off), "v"(gaddr) : "memory");
}
static __device__ __forceinline__ void wait_asynccnt0() {
  asm volatile("s_wait_asynccnt 0x0" ::: "memory");
}

// ---------------------------------------------------------------------------
// Kernel 1: attention convs (aq) and shifted doc masks (ad).
// aq/ad = 0 for invalid tail rows so padded feature rows never contribute.
// ---------------------------------------------------------------------------
__global__ void attn_mask_kernel(const int* __restrict__ qwt,
                                 const float* __restrict__ qwm,
                                 const float* __restrict__ dwm,
                                 const float* __restrict__ attn_table,
                                 const float* __restrict__ wa_uni, const float* __restrict__ ba_uni,
                                 const float* __restrict__ wa_bi,  const float* __restrict__ ba_bi,
                                 const float* __restrict__ wa_tri, const float* __restrict__ ba_tri,
                                 float* __restrict__ aq, float* __restrict__ ad) {
  const int b = blockIdx.x;
  const int tid = threadIdx.x;
  if (tid < LQ) {
    const int l = tid;
    const float a0 = attn_table[qwt[b * LQ + l]];
    const float a1 = (l + 1 < LQ) ? attn_table[qwt[b * LQ + l + 1]] : 0.f;
    const float a2 = (l + 2 < LQ) ? attn_table[qwt[b * LQ + l + 2]] : 0.f;
    const float m = qwm[b * LQ + l];
    float* aqb = aq + (size_t)b * 3 * LQ;
    aqb[0 * LQ + l] = m * fmaxf(wa_uni[0] * a0 + ba_uni[0], 0.f);
    aqb[1 * LQ + l] = (l < LQ - 1)
        ? m * fmaxf(wa_bi[0] * a0 + wa_bi[1] * a1 + ba_bi[0], 0.f) : 0.f;
    aqb[2 * LQ + l] = (l < LQ - 2)
        ? m * fmaxf(wa_tri[0] * a0 + wa_tri[1] * a1 + wa_tri[2] * a2 + ba_tri[0], 0.f) : 0.f;
  }
  for (int l = tid; l < LD; l += blockDim.x) {
    const float m = dwm[b * LD + l];
    float* adb = ad + (size_t)b * 3 * LD;
    adb[0 * LD + l] = m;
    adb[1 * LD + l] = (l < LD - 1) ? m : 0.f;
    adb[2 * LD + l] = (l < LD - 2) ? m : 0.f;
  }
}

// ---------------------------------------------------------------------------
// Kernel 2: gather + n-gram conv (WMMA) + ReLU + eps + L2-normalize.
// One block per (16-position tile, batch); 8 waves, wave w owns C columns
// [16w,16w+16). n-gram taps = row-shifted A operands. Six fully independent
// WMMA accumulator chains (u, b0, b1, t0, t1, t2) keep the matrix pipe busy.
// Embedding gather goes global->LDS via async B128 copies.
// ---------------------------------------------------------------------------
__global__ void __launch_bounds__(256) feat_kernel(
    const float* __restrict__ emb, const int* __restrict__ toks, int L,
    const float* __restrict__ Wu, const float* __restrict__ Wb,
    const float* __restrict__ Wt,
    const float* __restrict__ bu, const float* __restrict__ bb,
    const float* __restrict__ bt,
    float* __restrict__ out) {
  __shared__ int   sm_tok[18];
  __shared__ float sm_emb[18 * EPAD];          // 22.2 KB
  __shared__ float sm_out[3 * 16 * OPAD];      // 25.3 KB
  __shared__ float sm_inv[48];

  const int tid = threadIdx.x;
  const int b = blockIdx.y;
  const int l0 = blockIdx.x * 16;

  if (tid < 18) {
    int l = l0 + tid;
    if (l > L - 1) l = L - 1;                  // clamp: tail rows masked later
    sm_tok[tid] = toks[(size_t)b * L + l];
  }
  __syncthreads();
  // 18 rows x 300 floats = 1350 x 16B async transfers, emb rows are 16B-aligned
  for (int i = tid; i < 18 * (D_EMB / 4); i += 256) {
    const int row = i / (D_EMB / 4);
    const int c4 = i - row * (D_EMB / 4);
    const unsigned lds_off =
        (unsigned)(uintptr_t)&sm_emb[row * EPAD + c4 * 4];
    const unsigned long long ga =
        (unsigned long long)(uintptr_t)(emb + (size_t)sm_tok[row] * D_EMB + c4 * 4);
    async_copy_b128(lds_off, ga);
  }
  wait_asynccnt0();
  __syncthreads();

  const int lane = tid & 31;
  const int w = tid >> 5;
  const int c0 = w * 16;
  const int n = lane & 15;
  const int h = lane >> 4;
  const int h2 = h * 2;

  v8f au  = {0, 0, 0, 0, 0, 0, 0, 0};
  v8f ab0 = {0, 0, 0, 0, 0, 0, 0, 0};
  v8f ab1 = {0, 0, 0, 0, 0, 0, 0, 0};
  v8f at0 = {0, 0, 0, 0, 0, 0, 0, 0};
  v8f at1 = {0, 0, 0, 0, 0, 0, 0, 0};
  v8f at2 = {0, 0, 0, 0, 0, 0, 0, 0};

  for (int k0 = 0; k0 < D_EMB; k0 += 4) {
    const float* pa = &sm_emb[n * EPAD + k0 + h2];
    v2f a0 = {pa[0], pa[1]};                   // shift 0
    v2f a1 = {pa[EPAD], pa[EPAD + 1]};         // shift 1
    v2f a2 = {pa[2 * EPAD], pa[2 * EPAD + 1]}; // shift 2
    const int koff = (k0 + h2) * C_CH + c0 + n;
    v2f fbu  = {Wu[koff],            Wu[koff + C_CH]};
    v2f fbb0 = {Wb[koff],            Wb[koff + C_CH]};
    v2f fbb1 = {Wb[W1 + koff],       Wb[W1 + koff + C_CH]};
    v2f fbt0 = {Wt[koff],            Wt[koff + C_CH]};
    v2f fbt1 = {Wt[W1 + koff],       Wt[W1 + koff + C_CH]};
    v2f fbt2 = {Wt[2 * W1 + koff],   Wt[2 * W1 + koff + C_CH]};
    au  = wmma4(a0, fbu,  au);                 // 6 independent chains
    ab0 = wmma4(a0, fbb0, ab0);
    ab1 = wmma4(a1, fbb1, ab1);
    at0 = wmma4(a0, fbt0, at0);
    at1 = wmma4(a1, fbt1, at1);
    at2 = wmma4(a2, fbt2, at2);
  }
  const v8f accb = ab0 + ab1;
  const v8f acct = at0 + at1 + at2;

  const float bvu = bu[c0 + n];
  const float bvb = bb[c0 + n];
  const float bvt = bt[c0 + n];
#pragma unroll
  for (int r = 0; r < 8; ++r) {
    const int M = r + h * 8;
    sm_out[(0 * 16 + M) * OPAD + c0 + n] = fmaxf(au[r]   + bvu, 0.f) + 1e-9f;
    sm_out[(1 * 16 + M) * OPAD + c0 + n] = fmaxf(accb[r] + bvb, 0.f) + 1e-9f;
    sm_out[(2 * 16 + M) * OPAD + c0 + n] = fmaxf(acct[r] + bvt, 0.f) + 1e-9f;
  }
  __syncthreads();

  if (tid < 48) {                              // row = g*16 + M = tid
    const float* row = &sm_out[tid * OPAD];
    float ss = 0.f;
    for (int c = 0; c < C_CH; ++c) { const float v = row[c]; ss += v * v; }
    sm_inv[tid] = 1.f / fmaxf(sqrtf(ss), 1e-10f);
  }
  __syncthreads();

  const int Lpad = gridDim.x * 16;             // 32 for q, 512 for d
  for (int i = tid; i < 3 * 16 * C_CH; i += 256) {
    const int g = i >> 11;
    const int rem = i & 2047;
    const int M = rem >> 7;
    const int c = rem & 127;
    out[(((size_t)b * 3 + g) * Lpad + (l0 + M)) * C_CH + c] =
        sm_out[(g * 16 + M) * OPAD + c] * sm_inv[g * 16 + M];
  }
}

// ---------------------------------------------------------------------------
// Kernel 3: similarity GEMM (WMMA) + 11-bin RBF pooling + log-pool.
// One block per (pool-pair p, batch). Doc 64x128 tiles are fetched by the
// Tensor Data Mover: 2D D# with pad_enable (128 dwords -> +4 dwords) so LDS
// rows land at the 132-float conflict-free stride; one wave issues, all waves
// sync behind s_wait_tensorcnt + barrier. Split-K dual accumulators break the
// serial WMMA RAW chain.
// ---------------------------------------------------------------------------
__global__ void __launch_bounds__(256) pool_kernel(
    const float* __restrict__ qfeat, const float* __restrict__ dfeat,
    const float* __restrict__ aq, const float* __restrict__ ad,
    const float* __restrict__ mu, const float* __restrict__ sigma,
    float* __restrict__ pools) {
  __shared__ float sm_q[LQ * QPAD];            // 16.9 KB
  __shared__ float sm_d[64 * QPAD];            // 33.8 KB
  __shared__ float sm_ps[LQ * NB];             // 1.4 KB

  const int p = blockIdx.x;
  const int b = blockIdx.y;
  const int tid = threadIdx.x;

  // pool order: (u,u),(u,t),(u,b),(b,u),(t,u),(b,b),(b,t),(t,b),(t,t)
  const int qg_tab[9] = {0, 0, 0, 1, 2, 1, 1, 2, 2};
  const int dg_tab[9] = {0, 2, 1, 0, 0, 1, 2, 1, 2};
  const int qg = qg_tab[p];
  const int dg = dg_tab[p];

  // q tile (32x128) via async B128 copies into padded LDS
  const float* qsrc = qfeat + ((size_t)b * 3 + qg) * LQ * C_CH;
  for (int i = tid; i < LQ * C_CH / 4; i += 256) {
    const int row = i >> 5;
    const int c4 = i & 31;
    async_copy_b128((unsigned)(uintptr_t)&sm_q[row * QPAD + c4 * 4],
                    (unsigned long long)(uintptr_t)(qsrc + row * C_CH + c4 * 4));
  }
  for (int i = tid; i < LQ * NB; i += 256) sm_ps[i] = 0.f;

  float muv[NB], ni2[NB];
#pragma unroll
  for (int i = 0; i < NB; ++i) {
    muv[i] = mu[i];
    const float s = sigma[i];
    ni2[i] = -0.5f / (s * s);
  }

  const int lane = tid & 31;
  const int w = tid >> 5;
  const int m = w >> 2;                        // M-tile (q rows 0-15 / 16-31)
  const int nt = w & 3;                        // 16-wide slice of the 64-chunk
  const int n = lane & 15;
  const int h = lane >> 4;
  const int h2 = h * 2;

  float ps_lane[8][NB];
#pragma unroll
  for (int r = 0; r < 8; ++r)
#pragma unroll
    for (int k = 0; k < NB; ++k) ps_lane[r][k] = 0.f;

  const float* dsrc = dfeat + ((size_t)b * 3 + dg) * LD * C_CH;
  const float* adb = ad + ((size_t)b * 3 + dg) * LD;

  // TDM descriptor pieces that do not change per chunk (all wave-uniform).
  // group1: data_size=4B(2), pad_enable, pad_interval=128dw(6), pad_amount=4dw(3)
  u32x8 g1;
  g1[0] = 0x07920000u;
  g1[1] = ((unsigned)C_CH) << 16;              // tensor_dim0 = 128 (lo16)
  g1[2] = ((unsigned)LD) << 16;                // tensor_dim1 = 512 (lo16)
  g1[3] = ((unsigned)C_CH) << 16;              // tile_dim0   = 128
  g1[4] = 64u;                                 // tile_dim1   = 64
  g1[5] = (unsigned)C_CH;                      // tensor_dim0_stride = 128
  g1[6] = 0u;
  g1[7] = 0u;
  const u32x4 gz = {0u, 0u, 0u, 0u};
  const unsigned lds_d = (unsigned)(uintptr_t)&sm_d[0];

  wait_asynccnt0();                            // q tile resident
  __syncthreads();

  for (int chunk = 0; chunk < LD / 64; ++chunk) {
    __syncthreads();                           // previous chunk fully consumed
    if (w == 0) {
      const unsigned long long ga =
          (unsigned long long)(uintptr_t)(dsrc + (size_t)(chunk * 64) * C_CH);
      u32x4 g0;
      g0[0] = 1u;                              // count=1, user descriptor
      g0[1] = lds_d;                           // LDS byte address
      g0[2] = (unsigned)(ga & 0xFFFFFFFFull);  // global_addr[31:0]
      g0[3] = (unsigned)((ga >> 32) & 0x1FFFFFFu) | 0x80000000u; // addr[56:32]|type=2
      asm volatile("tensor_load_to_lds %0, %1, %2, %3"
                   :: "s"(g0), "s"(g1), "s"(gz), "s"(gz) : "memory");
      __builtin_amdgcn_s_wait_tensorcnt((short)0);
    }
    __syncthreads();

    v8f acc0 = {0, 0, 0, 0, 0, 0, 0, 0};       // split-K: two independent
    v8f acc1 = {0, 0, 0, 0, 0, 0, 0, 0};       // WMMA chains
    for (int k0 = 0; k0 < C_CH; k0 += 8) {
      const float* pq = &sm_q[(m * 16 + n) * QPAD + k0 + h2];
      const float* pd = &sm_d[(nt * 16 + n) * QPAD + k0 + h2];
      v2f a0 = {pq[0], pq[1]};
      v2f b0 = {pd[0], pd[1]};
      v2f a1 = {pq[4], pq[5]};
      v2f b1 = {pd[4], pd[5]};
      acc0 = wmma4(a0, b0, acc0);
      acc1 = wmma4(a1, b1, acc1);
    }

    const float adv = adb[chunk * 64 + nt * 16 + n];
#pragma unroll
    for (int r = 0; r < 8; ++r) {
      const float s = acc0[r] + acc1[r];
#pragma unroll
      for (int k = 0; k < NB; ++k) {
        const float d0 = s - muv[k];
        ps_lane[r][k] += adv * __expf(d0 * d0 * ni2[k]);
      }
    }
  }

  // reduce the 16 columns held by each lane-half, then one atomic per half
#pragma unroll
  for (int r = 0; r < 8; ++r) {
#pragma unroll
    for (int k = 0; k < NB; ++k) {
      float v = ps_lane[r][k];
      v += __shfl_xor(v, 1, 32);
      v += __shfl_xor(v, 2, 32);
      v += __shfl_xor(v, 4, 32);
      v += __shfl_xor(v, 8, 32);
      if (n == 0)
        atomicAdd(&sm_ps[(m * 16 + r + h * 8) * NB + k], v);
    }
  }
  __syncthreads();

  if (tid < NB) {
    const float* aqb = aq + ((size_t)b * 3 + qg) * LQ;
    float tot = 0.f;
    for (int lq = 0; lq < LQ; ++lq)
      tot += __logf(fmaxf(sm_ps[lq * NB + tid], 1e-10f)) * 0.01f * aqb[lq];
    pools[((size_t)b * 9 + p) * NB + tid] = tot;
  }
}

// ---------------------------------------------------------------------------
// Kernel 4: final dense + tanh.
// ---------------------------------------------------------------------------
__global__ void dense_kernel(const float* __restrict__ pools,
                             const float* __restrict__ Wd,
                             const float* __restrict__ bd,
                             float* __restrict__ out, int B) {
  const int b = blockIdx.x * blockDim.x + threadIdx.x;
  if (b >= B) return;
  float acc = bd[0];
  for (int j = 0; j < 9 * NB; ++j) acc += pools[(size_t)b * 9 * NB + j] * Wd[j];
  out[b] = tanhf(acc);
}

// ---------------------------------------------------------------------------
extern "C" void kernel_launch(void* const* d_in, const int* in_sizes, int n_in,
                              void* d_out, int out_size, void* d_ws, size_t ws_size,
                              hipStream_t stream) {
  const float* emb      = (const float*)d_in[0];
  const float* attn_tbl = (const float*)d_in[1];
  const float* W_uni    = (const float*)d_in[2];
  const float* b_uni    = (const float*)d_in[3];
  const float* W_bi     = (const float*)d_in[4];
  const float* b_bi     = (const float*)d_in[5];
  const float* W_tri    = (const float*)d_in[6];
  const float* b_tri    = (const float*)d_in[7];
  const float* wa_uni   = (const float*)d_in[8];
  const float* ba_uni   = (const float*)d_in[9];
  const float* wa_bi    = (const float*)d_in[10];
  const float* ba_bi    = (const float*)d_in[11];
  const float* wa_tri   = (const float*)d_in[12];
  const float* ba_tri   = (const float*)d_in[13];
  const float* W_dense  = (const float*)d_in[14];
  const float* b_dense  = (const float*)d_in[15];
  const float* mu       = (const float*)d_in[16];
  const float* sigma    = (const float*)d_in[17];
  const int*   qwt      = (const int*)d_in[18];
  const int*   dwt      = (const int*)d_in[19];
  const float* qwm      = (const float*)d_in[20];
  const float* dwm      = (const float*)d_in[21];

  const int B = in_sizes[18] / LQ;

  float* ws = (float*)d_ws;
  size_t off = 0;
  float* qfeat = ws + off; off += (size_t)B * 3 * LQ * C_CH;
  float* dfeat = ws + off; off += (size_t)B * 3 * LD * C_CH;
  float* aq    = ws + off; off += (size_t)B * 3 * LQ;
  float* ad    = ws + off; off += (size_t)B * 3 * LD;
  float* pools = ws + off;

  attn_mask_kernel<<<dim3(B), dim3(256), 0, stream>>>(
      qwt, qwm, dwm, attn_tbl, wa_uni, ba_uni, wa_bi, ba_bi, wa_tri, ba_tri,
      aq, ad);
  feat_kernel<<<dim3(LQ / 16, B), dim3(256), 0, stream>>>(
      emb, qwt, LQ, W_uni, W_bi, W_tri, b_uni, b_bi, b_tri, qfeat);
  feat_kernel<<<dim3(LD / 16, B), dim3(256), 0, stream>>>(
      emb, dwt, LD, W_uni, W_bi, W_tri, b_uni, b_bi, b_tri, dfeat);
  pool_kernel<<<dim3(9, B), dim3(256), 0, stream>>>(
      qfeat, dfeat, aq, ad, mu, sigma, pools);
  dense_kernel<<<dim3((B + 127) / 128), dim3(128), 0, stream>>>(
      pools, W_dense, b_dense, (float*)d_out, B);
}